// ChamferDistancePy_30442728194567
// MI455X (gfx1250) — compile-verified
//
#include <hip/hip_runtime.h>

typedef __attribute__((ext_vector_type(2))) float v2f;
typedef __attribute__((ext_vector_type(8))) float v8f;

__global__ void chamfer_zero_kernel(float* out) {
    if (threadIdx.x == 0) out[0] = 0.0f;
}

// One wave handles a 16-row strip of P and sweeps all columns of Q.
// d2(m,n) via one V_WMMA_F32_16X16X4_F32 per 16x16 tile:
//   A_m = (-2x,-2y,-2z,1)   [loop-invariant]
//   B_n = (x',y',z',|b|^2)  [per column tile]
//   C[m][n] = |a_m|^2       [loop-invariant, built once via shuffles]
// D = A*B + C = |a|^2 + |b|^2 - 2 a.b = d^2
__global__ __launch_bounds__(256) void chamfer_dir_kernel(
    const float* __restrict__ P, const float* __restrict__ Q,
    int nP, int nQ, float* __restrict__ out)
{
    const int lane = threadIdx.x & 31;
    const int wave = blockIdx.x * (blockDim.x >> 5) + (threadIdx.x >> 5);
    const int rowBase = wave * 16;
    if (rowBase >= nP) return;               // wave-uniform: EXEC stays all-1s

    const int  l16 = lane & 15;
    const bool hi  = lane >= 16;

    // ---- A-matrix setup (16x4 f32, 2 VGPRs per lane), loop-invariant ----
    // lanes 0-15: K=0,1 -> (-2x, -2y); lanes 16-31: K=2,3 -> (-2z, 1)
    const int m = rowBase + l16;
    const float ax = P[3 * m + 0];
    const float ay = P[3 * m + 1];
    const float az = P[3 * m + 2];
    const float sq1 = ax * ax + ay * ay + az * az;
    v2f A;
    A.x = hi ? (-2.0f * az) : (-2.0f * ax);
    A.y = hi ? 1.0f         : (-2.0f * ay);

    // ---- C accumulator = |a_m|^2, loop-invariant ----
    // C VGPR r: lanes 0-15 hold M=r, lanes 16-31 hold M=r+8.
    // sq1 in lane l holds row rowBase+(l&15); broadcast with per-lane src index.
    v8f Cacc;
#pragma unroll
    for (int r = 0; r < 8; ++r)
        Cacc[r] = __shfl(sq1, (hi ? 8 : 0) + r, 32);

    v8f dmin;
#pragma unroll
    for (int r = 0; r < 8; ++r) dmin[r] = 3.4e38f;

    // ---- sweep column tiles: 1 WMMA + b96 load + ~5 VALU per tile ----
#pragma unroll 4
    for (int t = 0; t < nQ; t += 16) {
        const int n = t + l16;
        const float bx = Q[3 * n + 0];
        const float by = Q[3 * n + 1];
        const float bz = Q[3 * n + 2];
        const float sq2 = bx * bx + by * by + bz * bz;

        // B (4x16 f32): lanes 0-15: K=0,1 -> (x', y'); lanes 16-31: K=2,3 -> (z', |b|^2)
        v2f B;
        B.x = hi ? bz  : bx;
        B.y = hi ? sq2 : by;

        v8f D = __builtin_amdgcn_wmma_f32_16x16x4_f32(
            /*neg_a=*/false, A, /*neg_b=*/false, B,
            /*c_mod=*/(short)0, Cacc, /*reuse_a=*/false, /*reuse_b=*/false);

#pragma unroll
        for (int r = 0; r < 8; ++r) dmin[r] = fminf(dmin[r], D[r]);
    }

    // ---- min-reduce across the 16 lanes of each half (wave32) ----
    // xor masks 1,2,4,8 never cross the lane-16 boundary.
#pragma unroll
    for (int mask = 1; mask < 16; mask <<= 1) {
#pragma unroll
        for (int r = 0; r < 8; ++r)
            dmin[r] = fminf(dmin[r], __shfl_xor(dmin[r], mask, 32));
    }

    // lanes 0-15 hold row-mins for rows rowBase+0..7 in dmin[0..7],
    // lanes 16-31 hold rows rowBase+8..15.  sqrt is monotone -> min then sqrt.
    float s = 0.0f;
#pragma unroll
    for (int r = 0; r < 8; ++r) s += sqrtf(fmaxf(dmin[r], 0.0f));

    s += __shfl_xor(s, 16, 32);              // combine both halves
    if (lane == 0) atomicAdd(out, s);
}

extern "C" void kernel_launch(void* const* d_in, const int* in_sizes, int n_in,
                              void* d_out, int out_size, void* d_ws, size_t ws_size,
                              hipStream_t stream) {
    const float* p1 = (const float*)d_in[0];
    const float* p2 = (const float*)d_in[1];
    const int N = in_sizes[0] / 3;
    const int M = in_sizes[1] / 3;
    float* out = (float*)d_out;

    chamfer_zero_kernel<<<1, 32, 0, stream>>>(out);

    const dim3 block(256);                       // 8 waves per block
    const int wavesPerBlock = 256 / 32;

    // Pass 1: row mins of dist(p1, p2)  -> sum(min1)
    {
        const int strips = (N + 15) / 16;
        const dim3 grid((strips + wavesPerBlock - 1) / wavesPerBlock);
        chamfer_dir_kernel<<<grid, block, 0, stream>>>(p1, p2, N, M, out);
    }
    // Pass 2: col mins == row mins of dist(p2, p1) -> sum(min2)
    {
        const int strips = (M + 15) / 16;
        const dim3 grid((strips + wavesPerBlock - 1) / wavesPerBlock);
        chamfer_dir_kernel<<<grid, block, 0, stream>>>(p2, p1, M, N, out);
    }
}